// GAT_65223373357473
// MI455X (gfx1250) — compile-verified
//
#include <hip/hip_runtime.h>
#include <math.h>

// ---------------- problem constants ----------------
#define NNODES 50000
#define NEDGES 400000
#define DIN    128
#define DH     32
#define HEADS  10
#define HID    (HEADS * DH)   // 320
#define DOUT   32
#define NEG_SLOPE 0.2f
#define EPS_BN 1e-5f

typedef float v2f __attribute__((ext_vector_type(2)));
typedef float v8f __attribute__((ext_vector_type(8)));

// ---------------- helpers ----------------
__device__ __forceinline__ void atomicMaxFloat(float* addr, float val) {
    // Sign-aware ordered-int trick. Works when initialized to -INF.
    if (val >= 0.0f) {
        atomicMax((int*)addr, __float_as_int(val));
    } else {
        atomicMin((unsigned int*)addr, __float_as_uint(val));
    }
}

__device__ __forceinline__ float waveReduceSum(float v) {
    #pragma unroll
    for (int o = 16; o > 0; o >>= 1) v += __shfl_xor(v, o, 32);
    return v;
}

__device__ __forceinline__ float waveReduceMax(float v) {
    #pragma unroll
    for (int o = 16; o > 0; o >>= 1) v = fmaxf(v, __shfl_xor(v, o, 32));
    return v;
}

// ---------------- workspace init ----------------
__global__ void init_ws_kernel(float* agg1, float* segmax1, float* segsum1,
                               float* agg2, float* segmax2, float* segsum2,
                               float* bnstats) {
    int i = blockIdx.x * blockDim.x + threadIdx.x;
    if (i < NNODES * HID) agg1[i] = 0.0f;
    if (i < NNODES * HEADS) { segmax1[i] = -INFINITY; segsum1[i] = 0.0f; }
    if (i < NNODES * DOUT) agg2[i] = 0.0f;
    if (i < NNODES) { segmax2[i] = -INFINITY; segsum2[i] = 0.0f; }
    if (i < 1280) bnstats[i] = 0.0f;
}

// ---------------- f32 WMMA GEMM:  C[M,Nn] = A[M,K] @ B[K,Nn] + bias ----------------
// One wave computes one 16x16 tile using V_WMMA_F32_16X16X4_F32.
// ISA layouts (wave32):
//   A 16x4 : lane holds m = lane%16 ; VGPR j holds k = kb + j + 2*(lane/16)
//   B 4x16 : lane holds n = lane%16 ; VGPR j holds k = kb + j + 2*(lane/16)
//   C 16x16: lane holds n = lane%16 ; VGPR j holds m = j + 8*(lane/16)
__global__ void wmma_gemm_bias(const float* __restrict__ A,
                               const float* __restrict__ B,
                               const float* __restrict__ bias,
                               float* __restrict__ C,
                               int K, int Nn, int Mtiles) {
    const int wave  = threadIdx.x >> 5;
    const int lane  = threadIdx.x & 31;
    const int tileM = blockIdx.x * (blockDim.x >> 5) + wave;
    if (tileM >= Mtiles) return;            // wave-uniform: EXEC stays all-ones
    const int tileN = blockIdx.y;
    const int half  = lane >> 4;            // 0 or 1
    const int col   = lane & 15;
    const int m = tileM * 16 + col;
    const int n = tileN * 16 + col;

    const float* arow = A + (size_t)m * K + 2 * half;      // + j gives k = kb + j + 2*half
    const float* bcol = B + (size_t)(2 * half) * Nn + n;

    v8f acc = {};
    for (int kb = 0; kb < K; kb += 4) {
        v2f a = *(const v2f*)(arow + kb);                  // 8B-aligned (K, kb even)
        v2f b;
        b.x = bcol[(size_t)kb * Nn];
        b.y = bcol[(size_t)(kb + 1) * Nn];
        acc = __builtin_amdgcn_wmma_f32_16x16x4_f32(
                  /*neg_a=*/false, a, /*neg_b=*/false, b,
                  /*c_mod=*/(short)0, acc, /*reuse_a=*/false, /*reuse_b=*/false);
    }

    const float bn = bias[n];
    #pragma unroll
    for (int j = 0; j < 8; ++j) {
        C[(size_t)(tileM * 16 + j + 8 * half) * Nn + n] = acc[j] + bn;
    }
}

// ---------------- edge phase: score + segment max ----------------
// One wave per (edge, head); lane = channel (C = 32).
__global__ void edge_score_kernel(const float* __restrict__ xl,
                                  const float* __restrict__ xr,
                                  const float* __restrict__ att,
                                  const int* __restrict__ src,
                                  const int* __restrict__ dst,
                                  float* __restrict__ score,
                                  float* __restrict__ segmax,
                                  int E, int Hh) {
    const int w    = (blockIdx.x * blockDim.x + threadIdx.x) >> 5;
    const int lane = threadIdx.x & 31;
    if (w >= E * Hh) return;
    const int e = w / Hh;
    const int h = w - e * Hh;
    const int s = src[e];
    const int d = dst[e];
    const int HC = Hh * 32;

    float v = xl[(size_t)s * HC + h * 32 + lane] + xr[(size_t)d * HC + h * 32 + lane];
    v = (v >= 0.0f) ? v : NEG_SLOPE * v;                   // leaky_relu
    v *= att[h * 32 + lane];
    v = waveReduceSum(v);
    if (lane == 0) {
        score[(size_t)e * Hh + h] = v;
        atomicMaxFloat(&segmax[(size_t)d * Hh + h], v);
    }
}

// ---------------- edge phase: exp + segment sum ----------------
__global__ void edge_exp_kernel(const float* __restrict__ segmax,
                                const int* __restrict__ dst,
                                float* __restrict__ score,
                                float* __restrict__ segsum,
                                int E, int Hh) {
    const int i = blockIdx.x * blockDim.x + threadIdx.x;
    if (i >= E * Hh) return;
    const int e = i / Hh;
    const int h = i - e * Hh;
    const int d = dst[e];
    const float a = expf(score[i] - segmax[(size_t)d * Hh + h]);
    score[i] = a;                                          // reuse buffer for exp value
    atomicAdd(&segsum[(size_t)d * Hh + h], a);
}

// ---------------- edge phase: weighted aggregation ----------------
__global__ void edge_aggregate_kernel(const float* __restrict__ xl,
                                      const float* __restrict__ score,
                                      const float* __restrict__ segsum,
                                      const int* __restrict__ src,
                                      const int* __restrict__ dst,
                                      float* __restrict__ out,
                                      int E, int Hh) {
    const int w    = (blockIdx.x * blockDim.x + threadIdx.x) >> 5;
    const int lane = threadIdx.x & 31;
    if (w >= E * Hh) return;
    const int e = w / Hh;
    const int h = w - e * Hh;
    const int s = src[e];
    const int d = dst[e];
    const float alpha = score[(size_t)e * Hh + h] /
                        (segsum[(size_t)d * Hh + h] + 1e-16f);
    const int HC = Hh * 32;
    atomicAdd(&out[(size_t)d * HC + h * 32 + lane],
              xl[(size_t)s * HC + h * 32 + lane] * alpha);
}

// ---------------- BatchNorm statistics (biased variance, matches jnp.var) ----------
__global__ void bn_stats_kernel(const float* __restrict__ agg,
                                const float* __restrict__ bias1,
                                float* __restrict__ bnsum,
                                float* __restrict__ bnsq) {
    const int c  = threadIdx.x;              // 0..319
    const int r0 = blockIdx.x * 64;
    const float b = bias1[c];
    float s = 0.0f, ss = 0.0f;
    for (int r = 0; r < 64; ++r) {
        const int row = r0 + r;
        if (row < NNODES) {
            const float v = agg[(size_t)row * HID + c] + b;
            s += v; ss += v * v;
        }
    }
    atomicAdd(&bnsum[c], s);
    atomicAdd(&bnsq[c], ss);
}

__global__ void bn_finalize_kernel(const float* __restrict__ bnsum,
                                   const float* __restrict__ bnsq,
                                   const float* __restrict__ gamma,
                                   float* __restrict__ mu_out,
                                   float* __restrict__ sc_out) {
    const int c = threadIdx.x;
    if (c >= HID) return;
    const float invN = 1.0f / (float)NNODES;
    const float mu  = bnsum[c] * invN;
    const float var = bnsq[c] * invN - mu * mu;
    mu_out[c] = mu;
    sc_out[c] = rsqrtf(var + EPS_BN) * gamma[c];
}

__global__ void bn_apply_kernel(float* __restrict__ h,
                                const float* __restrict__ bias1,
                                const float* __restrict__ mu,
                                const float* __restrict__ sc,
                                const float* __restrict__ beta,
                                const float* __restrict__ prelu_a) {
    const int i = blockIdx.x * blockDim.x + threadIdx.x;
    if (i >= NNODES * HID) return;
    const int c = i % HID;
    const float a0 = prelu_a[0];
    float v = h[i] + bias1[c];
    v = (v - mu[c]) * sc[c] + beta[c];
    v = (v >= 0.0f) ? v : a0 * v;            // PReLU
    h[i] = v;                                // in place
}

// ---------------- final: +bias2, then log_softmax over 32 cols ----------------
__global__ void final_kernel(const float* __restrict__ agg2,
                             const float* __restrict__ bias2,
                             float* __restrict__ out,
                             float* __restrict__ lsm) {
    const int w    = (blockIdx.x * blockDim.x + threadIdx.x) >> 5;
    const int lane = threadIdx.x & 31;
    if (w >= NNODES) return;
    const float v = agg2[(size_t)w * DOUT + lane] + bias2[lane];
    out[(size_t)w * DOUT + lane] = v;
    const float mx = waveReduceMax(v);
    const float se = waveReduceSum(expf(v - mx));
    lsm[(size_t)w * DOUT + lane] = v - mx - logf(se);
}

// ---------------- host launcher ----------------
extern "C" void kernel_launch(void* const* d_in, const int* in_sizes, int n_in,
                              void* d_out, int out_size, void* d_ws, size_t ws_size,
                              hipStream_t stream) {
    (void)in_sizes; (void)n_in; (void)out_size; (void)ws_size;

    const float* x      = (const float*)d_in[0];
    const int*   ei     = (const int*)  d_in[1];
    const float* Wl1    = (const float*)d_in[2];
    const float* bl1    = (const float*)d_in[3];
    const float* Wr1    = (const float*)d_in[4];
    const float* br1    = (const float*)d_in[5];
    const float* att1   = (const float*)d_in[6];
    const float* bias1  = (const float*)d_in[7];
    const float* gamma  = (const float*)d_in[8];
    const float* beta   = (const float*)d_in[9];
    const float* prelua = (const float*)d_in[10];
    const float* Wl2    = (const float*)d_in[11];
    const float* bl2    = (const float*)d_in[12];
    const float* Wr2    = (const float*)d_in[13];
    const float* br2    = (const float*)d_in[14];
    const float* att2   = (const float*)d_in[15];
    const float* bias2  = (const float*)d_in[16];

    const int* src = ei;            // edge_index[0, :]
    const int* dst = ei + NEDGES;   // edge_index[1, :]

    // workspace layout (floats)
    float* ws      = (float*)d_ws;
    float* xl1     = ws;                          // N*HID   = 16,000,000
    float* xr1     = xl1     + (size_t)NNODES * HID;
    float* agg1    = xr1     + (size_t)NNODES * HID;      // aggregation, then h (in place)
    float* score1  = agg1    + (size_t)NNODES * HID;      // E*HEADS = 4,000,000
    float* segmax1 = score1  + (size_t)NEDGES * HEADS;    // N*HEADS
    float* segsum1 = segmax1 + (size_t)NNODES * HEADS;
    float* xl2     = segsum1 + (size_t)NNODES * HEADS;    // N*DOUT
    float* xr2     = xl2     + (size_t)NNODES * DOUT;
    float* score2  = xr2     + (size_t)NNODES * DOUT;     // E
    float* segmax2 = score2  + (size_t)NEDGES;            // N
    float* segsum2 = segmax2 + (size_t)NNODES;
    float* agg2    = segsum2 + (size_t)NNODES;            // N*DOUT
    float* bnstats = agg2    + (size_t)NNODES * DOUT;     // 4*320
    float* bnsum   = bnstats;
    float* bnsq    = bnstats + 320;
    float* bnmu    = bnstats + 640;
    float* bnsc    = bnstats + 960;

    float* out_f = (float*)d_out;                 // [N, 32] out
    float* lsm_f = out_f + (size_t)NNODES * DOUT; // [N, 32] log_softmax

    const int MT = NNODES / 16;                   // 3125, exact

    // 0) init accumulators / segment buffers
    {
        const int total = NNODES * HID;
        init_ws_kernel<<<(total + 255) / 256, 256, 0, stream>>>(
            agg1, segmax1, segsum1, agg2, segmax2, segsum2, bnstats);
    }

    // 1) layer-1 linear transforms (f32 WMMA)
    {
        dim3 grid((MT + 3) / 4, HID / 16);        // (782, 20)
        wmma_gemm_bias<<<grid, 128, 0, stream>>>(x, Wl1, bl1, xl1, DIN, HID, MT);
        wmma_gemm_bias<<<grid, 128, 0, stream>>>(x, Wr1, br1, xr1, DIN, HID, MT);
    }

    // 2) layer-1 edge softmax + aggregation
    {
        const int waves  = NEDGES * HEADS;        // 4,000,000
        const int blocks = (waves * 32 + 255) / 256;
        edge_score_kernel<<<blocks, 256, 0, stream>>>(
            xl1, xr1, att1, src, dst, score1, segmax1, NEDGES, HEADS);
        edge_exp_kernel<<<(waves + 255) / 256, 256, 0, stream>>>(
            segmax1, dst, score1, segsum1, NEDGES, HEADS);
        edge_aggregate_kernel<<<blocks, 256, 0, stream>>>(
            xl1, score1, segsum1, src, dst, agg1, NEDGES, HEADS);
    }

    // 3) BatchNorm (batch stats) + PReLU, in place on agg1 -> h
    {
        bn_stats_kernel<<<(NNODES + 63) / 64, HID, 0, stream>>>(agg1, bias1, bnsum, bnsq);
        bn_finalize_kernel<<<1, HID, 0, stream>>>(bnsum, bnsq, gamma, bnmu, bnsc);
        bn_apply_kernel<<<(NNODES * HID + 255) / 256, 256, 0, stream>>>(
            agg1, bias1, bnmu, bnsc, beta, prelua);
    }

    // 4) layer-2 linear transforms (f32 WMMA, K = 320)
    {
        dim3 grid((MT + 3) / 4, DOUT / 16);       // (782, 2)
        wmma_gemm_bias<<<grid, 128, 0, stream>>>(agg1, Wl2, bl2, xl2, HID, DOUT, MT);
        wmma_gemm_bias<<<grid, 128, 0, stream>>>(agg1, Wr2, br2, xr2, HID, DOUT, MT);
    }

    // 5) layer-2 edge softmax + aggregation (H = 1)
    {
        const int waves  = NEDGES;
        const int blocks = (waves * 32 + 255) / 256;
        edge_score_kernel<<<blocks, 256, 0, stream>>>(
            xl2, xr2, att2, src, dst, score2, segmax2, NEDGES, 1);
        edge_exp_kernel<<<(waves + 255) / 256, 256, 0, stream>>>(
            segmax2, dst, score2, segsum2, NEDGES, 1);
        edge_aggregate_kernel<<<blocks, 256, 0, stream>>>(
            xl2, score2, segsum2, src, dst, agg2, NEDGES, 1);
    }

    // 6) out = agg2 + bias2 ; log_softmax over the 32 columns
    {
        const int blocks = (NNODES * 32 + 255) / 256;
        final_kernel<<<blocks, 256, 0, stream>>>(agg2, bias2, out_f, lsm_f);
    }
}